// Gamba5_56942676411223
// MI455X (gfx1250) — compile-verified
//
#include <hip/hip_runtime.h>
#include <hip/hip_bf16.h>

typedef __attribute__((ext_vector_type(2))) float v2f;
typedef __attribute__((ext_vector_type(8))) float v8f;

// ---- problem constants ----
constexpr int E_ = 5, K_ = 6;
constexpr int B_ = 128, L_ = 1024, D_ = 768;
constexpr int DS_ = 128, DC_ = 4, HD_ = 64;
constexpr int DI_ = 2 * D_;              // 1536
constexpr int H_ = DI_ / HD_;            // 24
constexpr int CONV_ = DI_ + 2 * DS_;     // 1792
constexpr int DPROJ_ = 2 * DI_ + 2 * DS_ + H_; // 3352
constexpr int NPAD_ = 3360;              // 3352 padded to mult of 32
constexpr int F_ = 4 * D_;               // 3072
constexpr int NL_ = 3;

// ---- device math helpers ----
__device__ __forceinline__ float silu_f(float x) { return x / (1.0f + __expf(-x)); }
__device__ __forceinline__ float softplus_f(float x) {
    return (x > 20.0f) ? x : log1pf(expf(x));
}
__device__ __forceinline__ float gelu_f(float x) {
    return 0.5f * x * (1.0f + erff(x * 0.70710678118654752440f));
}

__device__ __forceinline__ float block_reduce256(float v, float* red) {
    int tid = threadIdx.x;
    red[tid] = v;
    __syncthreads();
    #pragma unroll
    for (int s = 128; s > 0; s >>= 1) {
        if (tid < s) red[tid] += red[tid + s];
        __syncthreads();
    }
    float r = red[0];
    __syncthreads();
    return r;
}

// ---- 1) masked mean pool: (B,L,D) -> (B,D) ----
__global__ __launch_bounds__(256) void pool_kernel(const float* __restrict__ emb,
                                                   const float* __restrict__ mask,
                                                   float* __restrict__ pooled) {
    __shared__ float red[256];
    const int b = blockIdx.x, tid = threadIdx.x;
    float ms = 0.f;
    for (int l = tid; l < L_; l += 256) ms += mask[(long)b * L_ + l];
    float msum = block_reduce256(ms, red);
    float inv = 1.0f / (msum + 1e-8f);
    float a0 = 0.f, a1 = 0.f, a2 = 0.f;
    const float* eb = emb + (long)b * L_ * D_;
    const float* mb = mask + (long)b * L_;
    for (int l = 0; l < L_; ++l) {
        float m = mb[l];
        if (m != 0.0f) {
            const float* er = eb + (long)l * D_;
            a0 += m * er[tid];
            a1 += m * er[tid + 256];
            a2 += m * er[tid + 512];
        }
    }
    pooled[(long)b * D_ + tid]       = a0 * inv;
    pooled[(long)b * D_ + tid + 256] = a1 * inv;
    pooled[(long)b * D_ + tid + 512] = a2 * inv;
}

// ---- 2) gate: argmax_e (pooled@gate_w + gate_b + gumbel) ----
__global__ __launch_bounds__(128) void gate_kernel(const float* __restrict__ pooled,
                                                   const float* __restrict__ gate_w,
                                                   const float* __restrict__ gate_b,
                                                   const float* __restrict__ gum,
                                                   int* __restrict__ sel) {
    int b = threadIdx.x;
    if (b >= B_) return;
    float acc[E_];
    #pragma unroll
    for (int e = 0; e < E_; ++e) acc[e] = 0.f;
    const float* pr = pooled + (long)b * D_;
    for (int d = 0; d < D_; ++d) {
        float p = pr[d];
        #pragma unroll
        for (int e = 0; e < E_; ++e) acc[e] += p * gate_w[(long)d * E_ + e];
    }
    float bestv = -1e30f; int bi = 0;
    #pragma unroll
    for (int e = 0; e < E_; ++e) {
        float v = acc[e] + gate_b[e] + gum[(long)b * E_ + e];
        if (v > bestv) { bestv = v; bi = e; }
    }
    sel[b] = bi;
}

// ---- 3) broadcast pooled -> x (E,B,D) ----
__global__ __launch_bounds__(256) void bcast_kernel(const float* __restrict__ pooled,
                                                    float* __restrict__ x) {
    long n = (long)E_ * B_ * D_;
    for (long i = (long)blockIdx.x * 256 + threadIdx.x; i < n; i += (long)gridDim.x * 256)
        x[i] = pooled[i % ((long)B_ * D_)];
}

// ---- 4) rmsnorm: out = w * x / sqrt(sum(x^2)+1e-6), one block per (e,b) row ----
__global__ __launch_bounds__(256) void rmsnorm_kernel(const float* __restrict__ x,
                                                      const float* __restrict__ w, // (E,K,D)
                                                      float* __restrict__ out, int kb) {
    __shared__ float red[256];
    const int rb = blockIdx.x;           // e*B + b
    const int e = rb / B_;
    const int tid = threadIdx.x;
    const float* xr = x + (long)rb * D_;
    const float* wr = w + ((long)e * K_ + kb) * D_;
    float s = 0.f;
    #pragma unroll
    for (int j = 0; j < 3; ++j) { float v = xr[tid + j * 256]; s += v * v; }
    float tot = block_reduce256(s, red);
    float inv = 1.0f / sqrtf(tot + 1e-6f);
    #pragma unroll
    for (int j = 0; j < 3; ++j) {
        int d = tid + j * 256;
        out[(long)rb * D_ + d] = wr[d] * xr[d] * inv;
    }
}

// ---- 5) fp32 WMMA GEMM: C (+)= act(A @ W + bias), per-expert batched via blockIdx.z ----
// TM=128 => gridDim.y==1 for B=128, so each weight element is streamed exactly once.
// W tile stored K-pair interleaved in LDS so each B fragment is one aligned ds_load_b64.
constexpr int TM = 128, TN = 64, KC = 32;
constexpr int LDA = 34;          // A stride: even (8B-aligned frags), conflict-free
constexpr int LDBP = 2 * TN + 32; // 160 floats per K-pair row: +32-bank shift between lane halves

template <bool ACCUM, bool GELU>
__global__ __launch_bounds__(256) void gemm_wmma_kernel(
    const float* __restrict__ A, long sA,
    const float* __restrict__ W, long sW,
    const float* __restrict__ bias, long sBias,
    float* __restrict__ C, long sC,
    int M, int N, int ldC, int Kd) {
    __shared__ float As[TM * LDA];
    __shared__ float Wp[(KC / 2) * LDBP];
    const int e = blockIdx.z;
    A += (long)e * sA;
    W += (long)e * sW;
    C += (long)e * sC;
    const float* biasE = bias ? (bias + (long)e * sBias) : nullptr;

    const int m0 = blockIdx.y * TM;
    const int n0 = blockIdx.x * TN;
    const int tid = threadIdx.x;
    const int lane = tid & 31, wave = tid >> 5;
    const int l15 = lane & 15, kl = lane >> 4;
    const int wm = wave;                 // 0..7 : wave owns a full 16-row strip

    v8f acc[4];
    #pragma unroll
    for (int t = 0; t < 4; ++t) acc[t] = (v8f){};

    for (int k0 = 0; k0 < Kd; k0 += KC) {
        __syncthreads();
        // stage A: TM x KC (128x32 = 4096 floats, 16 per thread)
        #pragma unroll
        for (int i = 0; i < (TM * KC) / 256; ++i) {
            int idx = tid + i * 256;
            int r = idx >> 5, c = idx & 31;
            int gm = m0 + r;
            As[r * LDA + c] = (gm < M) ? A[(long)gm * Kd + k0 + c] : 0.0f;
        }
        // stage W: KC x TN, K-pair interleaved: Wp[(r/2)*LDBP + c*2 + (r&1)]
        #pragma unroll
        for (int i = 0; i < (KC * TN) / 256; ++i) {
            int idx = tid + i * 256;
            int r = idx >> 6, c = idx & 63;
            int gn = n0 + c;
            Wp[(r >> 1) * LDBP + c * 2 + (r & 1)] =
                (gn < N) ? W[(long)(k0 + r) * N + gn] : 0.0f;
            if (k0 + KC < Kd && gn < N)
                __builtin_prefetch(&W[(long)(k0 + KC + r) * N + gn], 0, 1);
        }
        __syncthreads();
        const float* ap = &As[(wm * 16 + l15) * LDA + 2 * kl];
        #pragma unroll
        for (int kk = 0; kk < KC; kk += 4) {
            v2f a = *(const v2f*)(ap + kk);            // A frag: VGPR0=K{0,2}, VGPR1=K{1,3}
            // krow = kk + 2*kl is even; fragment = (W[krow][n], W[krow+1][n]) adjacent in LDS
            const float* bp = &Wp[(kk / 2 + kl) * LDBP + l15 * 2];
            #pragma unroll
            for (int t = 0; t < 4; ++t) {
                v2f b = *(const v2f*)(bp + t * 32);
                acc[t] = __builtin_amdgcn_wmma_f32_16x16x4_f32(false, a, false, b, (short)0, acc[t], false, false);
            }
        }
    }
    // store: VGPR j = row (wm*16 + j + 8*kl), col = n strip + l15
    const int row_base = m0 + wm * 16 + 8 * kl;
    #pragma unroll
    for (int t = 0; t < 4; ++t) {
        int col = n0 + t * 16 + l15;
        if (col < N) {
            float bv = biasE ? biasE[col] : 0.0f;
            #pragma unroll
            for (int j = 0; j < 8; ++j) {
                int row = row_base + j;
                if (row < M) {
                    float v = acc[t][j] + bv;
                    if (GELU) v = gelu_f(v);
                    float* cp = &C[(long)row * ldC + col];
                    if (ACCUM) v += *cp;
                    *cp = v;
                }
            }
        }
    }
}

// ---- 6) SSM elementwise, one block per (e,b) row ----
__global__ __launch_bounds__(256) void ssm_kernel(const float* __restrict__ zx,   // (E*B, NPAD)
                                                  const float* __restrict__ conv_w, // (E,K,CONV,DC)
                                                  const float* __restrict__ conv_b, // (E,K,CONV)
                                                  const float* __restrict__ dt_bias,// (E,K,H)
                                                  const float* __restrict__ dss,    // (E,K,H)
                                                  const float* __restrict__ snw,    // (E,K,DI)
                                                  float* __restrict__ yln, int kb) {
    __shared__ float red[256];
    __shared__ float coef[H_];
    __shared__ float ytmp[DI_];
    const int rb = blockIdx.x;
    const int e = rb / B_;
    const int tid = threadIdx.x;
    const long ek = (long)e * K_ + kb;
    const float* row = zx + (long)rb * NPAD_;
    const float* cw = conv_w + ek * CONV_ * DC_;   // use [c*DC + DC-1]
    const float* cb = conv_b + ek * CONV_;
    // bc = sum_s silu_conv(Bm[s]) * silu_conv(Cm[s])
    float p = 0.f;
    for (int s = tid; s < DS_; s += 256) {
        int cB = DI_ + s, cC = DI_ + DS_ + s;
        float bv = silu_f(row[DI_ + cB] * cw[cB * DC_ + (DC_ - 1)] + cb[cB]);
        float cv = silu_f(row[DI_ + cC] * cw[cC * DC_ + (DC_ - 1)] + cb[cC]);
        p += bv * cv;
    }
    float bc = block_reduce256(p, red);
    if (tid < H_) {
        float dtv = softplus_f(row[DI_ + CONV_ + tid] + dt_bias[ek * H_ + tid]);
        coef[tid] = dtv * bc + dss[ek * H_ + tid];
    }
    __syncthreads();
    float ss = 0.f;
    #pragma unroll
    for (int j = 0; j < DI_ / 256; ++j) {
        int i = tid + j * 256;
        float xa = silu_f(row[DI_ + i] * cw[i * DC_ + (DC_ - 1)] + cb[i]);
        float yv = coef[i >> 6] * xa * silu_f(row[i]);
        ytmp[i] = yv;
        ss += yv * yv;
    }
    float tot = block_reduce256(ss, red);
    float scale = rsqrtf(tot / (float)DI_ + 1e-5f);
    const float* snr = snw + ek * DI_;
    #pragma unroll
    for (int j = 0; j < DI_ / 256; ++j) {
        int i = tid + j * 256;
        yln[(long)rb * DI_ + i] = ytmp[i] * scale * snr[i];
    }
}

// ---- 7) final: out[b] = x[sel[b], b, :] @ fc_w + fc_b ----
__global__ __launch_bounds__(256) void final_kernel(const float* __restrict__ x,
                                                    const int* __restrict__ sel,
                                                    const float* __restrict__ fcw,
                                                    const float* __restrict__ fcb,
                                                    float* __restrict__ out) {
    __shared__ float red[256];
    const int b = blockIdx.x, tid = threadIdx.x;
    const int e = sel[b];
    const float* xr = x + ((long)e * B_ + b) * D_;
    float a[NL_] = {0.f, 0.f, 0.f};
    #pragma unroll
    for (int j = 0; j < 3; ++j) {
        int d = tid + j * 256;
        float v = xr[d];
        #pragma unroll
        for (int c = 0; c < NL_; ++c) a[c] += v * fcw[(long)d * NL_ + c];
    }
    #pragma unroll
    for (int c = 0; c < NL_; ++c) {
        float r = block_reduce256(a[c], red);
        if (tid == 0) out[(long)b * NL_ + c] = r + fcb[c];
    }
}

extern "C" void kernel_launch(void* const* d_in, const int* in_sizes, int n_in,
                              void* d_out, int out_size, void* d_ws, size_t ws_size,
                              hipStream_t stream) {
    const float* emb     = (const float*)d_in[0];
    const float* mask    = (const float*)d_in[1];
    const float* gum     = (const float*)d_in[2];
    const float* n1w     = (const float*)d_in[3];
    const float* n2w     = (const float*)d_in[4];
    const float* wip     = (const float*)d_in[5];
    const float* convw   = (const float*)d_in[6];
    const float* convb   = (const float*)d_in[7];
    const float* dtb     = (const float*)d_in[8];
    // d_in[9] = A_log: unused in forward (reference never reads it)
    const float* dss     = (const float*)d_in[10];
    const float* snw     = (const float*)d_in[11];
    const float* wop     = (const float*)d_in[12];
    const float* w1      = (const float*)d_in[13];
    const float* b1      = (const float*)d_in[14];
    const float* w2      = (const float*)d_in[15];
    const float* b2      = (const float*)d_in[16];
    const float* gatew   = (const float*)d_in[17];
    const float* gateb   = (const float*)d_in[18];
    const float* fcw     = (const float*)d_in[19];
    const float* fcb     = (const float*)d_in[20];
    float* out = (float*)d_out;

    // workspace layout (floats)
    float* ws = (float*)d_ws;
    float* pooled = ws;                               // B*D
    float* x      = pooled + (long)B_ * D_;           // E*B*D
    float* h      = x + (long)E_ * B_ * D_;           // E*B*D (h and h2)
    float* zx     = h + (long)E_ * B_ * D_;           // E*B*NPAD
    float* yln    = zx + (long)E_ * B_ * NPAD_;       // E*B*DI
    float* tbuf   = yln + (long)E_ * B_ * DI_;        // E*B*F
    int*   sel    = (int*)(tbuf + (long)E_ * B_ * F_);

    pool_kernel<<<B_, 256, 0, stream>>>(emb, mask, pooled);
    gate_kernel<<<1, 128, 0, stream>>>(pooled, gatew, gateb, gum, sel);
    bcast_kernel<<<480, 256, 0, stream>>>(pooled, x);

    const dim3 blk(256);
    for (int k = 0; k < K_; ++k) {
        // h = rmsnorm(x, norm1_w)
        rmsnorm_kernel<<<E_ * B_, 256, 0, stream>>>(x, n1w, h, k);
        // zx = h @ in_proj_w[e,k]  (M=B, N=DPROJ, K=D)
        gemm_wmma_kernel<false, false><<<dim3((DPROJ_ + TN - 1) / TN, (B_ + TM - 1) / TM, E_), blk, 0, stream>>>(
            h, (long)B_ * D_,
            wip + (long)k * D_ * DPROJ_, (long)K_ * D_ * DPROJ_,
            nullptr, 0,
            zx, (long)B_ * NPAD_, B_, DPROJ_, NPAD_, D_);
        // SSM elementwise -> yln
        ssm_kernel<<<E_ * B_, 256, 0, stream>>>(zx, convw, convb, dtb, dss, snw, yln, k);
        // x += yln @ out_proj_w[e,k]  (M=B, N=D, K=DI)
        gemm_wmma_kernel<true, false><<<dim3(D_ / TN, (B_ + TM - 1) / TM, E_), blk, 0, stream>>>(
            yln, (long)B_ * DI_,
            wop + (long)k * DI_ * D_, (long)K_ * DI_ * D_,
            nullptr, 0,
            x, (long)B_ * D_, B_, D_, D_, DI_);
        // h2 = rmsnorm(x, norm2_w)
        rmsnorm_kernel<<<E_ * B_, 256, 0, stream>>>(x, n2w, h, k);
        // t = gelu(h2 @ w1 + b1)  (M=B, N=F, K=D)
        gemm_wmma_kernel<false, true><<<dim3(F_ / TN, (B_ + TM - 1) / TM, E_), blk, 0, stream>>>(
            h, (long)B_ * D_,
            w1 + (long)k * D_ * F_, (long)K_ * D_ * F_,
            b1 + (long)k * F_, (long)K_ * F_,
            tbuf, (long)B_ * F_, B_, F_, F_, D_);
        // x += t @ w2 + b2  (M=B, N=D, K=F)
        gemm_wmma_kernel<true, false><<<dim3(D_ / TN, (B_ + TM - 1) / TM, E_), blk, 0, stream>>>(
            tbuf, (long)B_ * F_,
            w2 + (long)k * F_ * D_, (long)K_ * F_ * D_,
            b2 + (long)k * D_, (long)K_ * D_,
            x, (long)B_ * D_, B_, D_, D_, F_);
    }

    final_kernel<<<B_, 256, 0, stream>>>(x, sel, fcw, fcb, out);
}